// LieTransport_49873160241876
// MI455X (gfx1250) — compile-verified
//
#include <hip/hip_runtime.h>
#include <cstdint>

#define B_ 4
#define C_ 16
#define H_ 128
#define W_ 128
#define R_ 64

// One 256-thread block per (b, h, w) pixel.
// Stage the 4 bilinear taps (4 x C x R floats = 16 KB) into LDS with
// CDNA5 async global->LDS DMA, then blend and write C*R floats.
__global__ __launch_bounds__(256) void LieTransport_warp_kernel(
    const float* __restrict__ h_prev,
    const float* __restrict__ flow,
    const float* __restrict__ dt,
    float* __restrict__ out)
{
    __shared__ float lds[4 * C_ * R_];   // 4096 floats = 16 KB

    const int pix = blockIdx.x;
    const int b   = pix >> 14;           // / (H*W)
    const int hw  = pix & 16383;
    const int h   = hw >> 7;
    const int w   = hw & 127;
    const int tid = threadIdx.x;

    // ---- sample coordinate (uniform per block; cheap redundant compute) ----
    const float dtb = dt[b];
    const size_t fbase = (size_t)b * (2 * H_ * W_) + (size_t)h * W_ + w;
    const float fx = flow[fbase];                 // x-component
    const float fy = flow[fbase + H_ * W_];       // y-component

    const float gx = -1.0f + (float)w * (2.0f / 127.0f) - fx * dtb;
    const float gy = -1.0f + (float)h * (2.0f / 127.0f) - fy * dtb;

    // unnormalize (align_corners=False) + border clip
    float x = ((gx + 1.0f) * 128.0f - 1.0f) * 0.5f;
    float y = ((gy + 1.0f) * 128.0f - 1.0f) * 0.5f;
    x = fminf(fmaxf(x, 0.0f), 127.0f);
    y = fminf(fmaxf(y, 0.0f), 127.0f);

    const float x0f = floorf(x), y0f = floorf(y);
    const float wx = x - x0f, wy = y - y0f;
    const int x0 = (int)x0f, y0 = (int)y0f;
    const int x1 = min(x0 + 1, 127), y1 = min(y0 + 1, 127);

    // ---- stage 4 taps into LDS via async global->LDS (ASYNCcnt path) ----
    const float* srcb = h_prev + (size_t)b * (C_ * H_ * W_ * R_);
    const int tap  = tid >> 6;           // 4 taps x 64 threads
    const int lane = tid & 63;
    const int ty = (tap & 2) ? y1 : y0;
    const int tx = (tap & 1) ? x1 : x0;
    const float* tapbase = srcb + ((size_t)ty * W_ + tx) * R_;   // c = 0
    const uint32_t lds_base = (uint32_t)(uintptr_t)(&lds[0]);

    #pragma unroll
    for (int i = 0; i < 4; ++i) {
        const int idx = i * 64 + lane;   // float4 index within tap [0, 256)
        const int c   = idx >> 4;
        const int r4  = idx & 15;
        const unsigned long long g =
            (unsigned long long)(uintptr_t)(tapbase + (size_t)c * (H_ * W_ * R_) + r4 * 4);
        const uint32_t l = lds_base + (uint32_t)((tap * 1024 + c * 64 + r4 * 4) * 4);
        asm volatile("global_load_async_to_lds_b128 %0, %1, off"
                     :: "v"(l), "v"(g) : "memory");
    }
    asm volatile("s_wait_asynccnt 0x0" ::: "memory");
    __syncthreads();

    // ---- blend: one (c, r4) float4 per thread ----
    const int c  = tid >> 4;
    const int r4 = tid & 15;
    const int o  = c * R_ + r4 * 4;

    const float4 v00 = *(const float4*)&lds[0 * 1024 + o];
    const float4 v01 = *(const float4*)&lds[1 * 1024 + o];
    const float4 v10 = *(const float4*)&lds[2 * 1024 + o];
    const float4 v11 = *(const float4*)&lds[3 * 1024 + o];

    const float w00 = (1.0f - wx) * (1.0f - wy);
    const float w01 = wx * (1.0f - wy);
    const float w10 = (1.0f - wx) * wy;
    const float w11 = wx * wy;

    float4 r;
    r.x = fmaf(v00.x, w00, fmaf(v01.x, w01, fmaf(v10.x, w10, v11.x * w11)));
    r.y = fmaf(v00.y, w00, fmaf(v01.y, w01, fmaf(v10.y, w10, v11.y * w11)));
    r.z = fmaf(v00.z, w00, fmaf(v01.z, w01, fmaf(v10.z, w10, v11.z * w11)));
    r.w = fmaf(v00.w, w00, fmaf(v01.w, w01, fmaf(v10.w, w10, v11.w * w11)));

    float* op = out + (size_t)b * (C_ * H_ * W_ * R_)
                    + (size_t)c * (H_ * W_ * R_)
                    + ((size_t)h * W_ + w) * R_
                    + r4 * 4;
    *(float4*)op = r;
}

extern "C" void kernel_launch(void* const* d_in, const int* in_sizes, int n_in,
                              void* d_out, int out_size, void* d_ws, size_t ws_size,
                              hipStream_t stream) {
    const float* h_prev = (const float*)d_in[0];  // [4,16,128,128,64] f32
    const float* flow   = (const float*)d_in[1];  // [4,2,128,128]     f32
    const float* dt     = (const float*)d_in[2];  // [4]               f32
    float* out = (float*)d_out;                   // [4,16,128,128,64] f32

    dim3 grid(B_ * H_ * W_);   // 65536 blocks, one per pixel
    dim3 block(256);
    LieTransport_warp_kernel<<<grid, block, 0, stream>>>(h_prev, flow, dt, out);
}